// StructuredVariational_12945031431046
// MI455X (gfx1250) — compile-verified
//
#include <hip/hip_runtime.h>

// CDNA5 / gfx1250, wave32. Matrix op: V_WMMA_F32_16X16X4_F32 (full fp32 —
// the workload is HBM-bound at ~21 MB / 23.3 TB/s ≈ 0.9 us, so low-precision
// WMMA buys nothing; fp32 keeps exact parity with the reference dtype).

typedef __attribute__((ext_vector_type(2))) float v2f;
typedef __attribute__((ext_vector_type(8))) float v8f;

#define D_Z      64
#define D_Y      8
#define NBLK     2048
#define NS       128          // n_sample
#define D_TOTAL  16448
#define NY       16384        // NBLK * D_Y
#define ROWS_WG  64           // y-rows per workgroup
#define EPS_STR  68           // LDS row stride (words): 4n+2h bank pattern -> conflict-free b64
#define LYZ_STR  68

// ---------------------------------------------------------------------------
// Main kernel: out[s, 64+row] = m[64+row] + eps_z[s,:] . Lyz[row,:]
//                               + tril(Ly_block) . eps_yblock[s,:]
// GEMM part via WMMA f32 16x16x4: M=16 samples, N=16 rows, K=64 in 16 steps.
// ---------------------------------------------------------------------------
__global__ __launch_bounds__(256)
void sv_y_kernel(const float* __restrict__ m,
                 const float* __restrict__ ly_blocks,   // [2048 * 36] tril
                 const float* __restrict__ lyz,         // [16384 x 64] row-major
                 const float* __restrict__ eps,         // [128 x 16448]
                 float* __restrict__ out)               // [128 x 16448]
{
    __shared__ float s_epsz[NS * EPS_STR];        // eps[:, 0:64]   (34816 B)
    __shared__ float s_lyz [ROWS_WG * LYZ_STR];   // Lyz tile       (17408 B)
    __shared__ float s_ly  [ROWS_WG * 8];         // dense tril rows (2048 B)

    const int t  = threadIdx.x;
    const int r0 = blockIdx.x * ROWS_WG;          // first y-row of this WG

    // ---- stage eps_z: 128 rows x 64 floats, half a row per thread ----
    {
        const int s    = t >> 1;
        const int part = (t & 1) * 32;
        const float4* g = (const float4*)(eps + (size_t)s * D_TOTAL + part);
        float4*       l = (float4*)(s_epsz + s * EPS_STR + part);
#pragma unroll
        for (int i = 0; i < 8; ++i) l[i] = g[i];
    }
    // ---- stage Lyz tile: rows [r0, r0+64) are 16 KB contiguous in HBM ----
    {
        const int lr = t >> 2;
        const int q  = (t & 3) * 16;
        const float4* g = (const float4*)(lyz + (size_t)(r0 + lr) * D_Z + q);
        float4*       l = (float4*)(s_lyz + lr * LYZ_STR + q);
#pragma unroll
        for (int i = 0; i < 4; ++i) l[i] = g[i];
    }
    // ---- stage dense (zero-padded) Ly tril rows: 64 rows x 8 ----
#pragma unroll
    for (int e = t; e < ROWS_WG * 8; e += 256) {
        const int lr = e >> 3, c = e & 7;
        const int b  = (r0 + lr) >> 3;        // global 8x8 block index
        const int r  = lr & 7;                // row within block
        float v = 0.0f;
        if (c <= r) v = ly_blocks[b * 36 + (r * (r + 1)) / 2 + c];
        s_ly[e] = v;
    }
    __syncthreads();

    const int lane = t & 31;                  // wave32
    const int w    = t >> 5;                  // wave id = M-tile (samples 16w..16w+15)
    const int n    = lane & 15;               // A: M index | B: N index | D: N index
    const int h    = lane >> 4;               // lane half -> K pair / D row-half

    // Preload all 16 A fragments (held across the 4 N-tiles).
    // A layout (16x4 f32): lane holds M = lane&15; VGPR v holds K = 2*half + v.
    v2f a[16];
    {
        const float* abase = s_epsz + (w * 16 + n) * EPS_STR + 2 * h;
#pragma unroll
        for (int kk = 0; kk < 16; ++kk)
            a[kk] = *(const v2f*)(abase + 4 * kk);
    }

#pragma unroll
    for (int nt = 0; nt < 4; ++nt) {
        const int lr   = nt * 16 + n;         // local y-row handled by this lane
        const int grow = r0 + lr;             // global y-row
        const float mv = m[D_Z + grow];

        v8f acc;                              // C init = broadcast m (same row all samples)
#pragma unroll
        for (int j = 0; j < 8; ++j) acc[j] = mv;

        // B layout (4x16 f32): lane holds N = lane&15; VGPR v holds K = 2*half + v.
        const float* bbase = s_lyz + lr * LYZ_STR + 2 * h;
#pragma unroll
        for (int kk = 0; kk < 16; ++kk) {
            v2f b = *(const v2f*)(bbase + 4 * kk);
            acc = __builtin_amdgcn_wmma_f32_16x16x4_f32(
                      false, a[kk], false, b, (short)0, acc, false, false);
        }

        // Block-diagonal Ly contribution (dense-padded 8-wide dot) + store.
        // D layout: VGPR j <-> sample M = j + 8*half; lane N = row.
        const int colb = D_Z + r0 + (lr & ~7);        // eps column base of this block
        const float4 ly0 = *(const float4*)(s_ly + lr * 8);
        const float4 ly1 = *(const float4*)(s_ly + lr * 8 + 4);
#pragma unroll
        for (int j = 0; j < 8; ++j) {
            const int s = w * 16 + h * 8 + j;
            const float* ep = eps + (size_t)s * D_TOTAL + colb;
            const float4 e0 = *(const float4*)(ep);
            const float4 e1 = *(const float4*)(ep + 4);
            const float add = e0.x * ly0.x + e0.y * ly0.y + e0.z * ly0.z + e0.w * ly0.w
                            + e1.x * ly1.x + e1.y * ly1.y + e1.z * ly1.z + e1.w * ly1.w;
            out[(size_t)s * D_TOTAL + D_Z + grow] = acc[j] + add;
        }
    }
}

// ---------------------------------------------------------------------------
// z-rows: out[s, i<64] = m[i] + sum_{j<=i} Lz[i,j] * eps[s,j]. ~0.27 MFLOP.
// ---------------------------------------------------------------------------
__global__ __launch_bounds__(256)
void sv_z_kernel(const float* __restrict__ m,
                 const float* __restrict__ lz,          // [2080] tril
                 const float* __restrict__ eps,
                 float* __restrict__ out)
{
    const int idx = blockIdx.x * 256 + threadIdx.x;     // 0..8191
    const int s = idx >> 6;
    const int i = idx & 63;
    const float* lrow = lz + (i * (i + 1)) / 2;
    const float* erow = eps + (size_t)s * D_TOTAL;
    float acc = m[i];
    for (int j = 0; j <= i; ++j) acc += lrow[j] * erow[j];
    out[(size_t)s * D_TOTAL + i] = acc;
}

extern "C" void kernel_launch(void* const* d_in, const int* in_sizes, int n_in,
                              void* d_out, int out_size, void* d_ws, size_t ws_size,
                              hipStream_t stream)
{
    const float* m   = (const float*)d_in[0];   // [16448]
    const float* lz  = (const float*)d_in[1];   // [2080]
    const float* ly  = (const float*)d_in[2];   // [73728]
    const float* lyz = (const float*)d_in[3];   // [1048576]
    const float* eps = (const float*)d_in[4];   // [128*16448]
    float* out = (float*)d_out;                 // [128*16448]

    sv_y_kernel<<<NY / ROWS_WG, 256, 0, stream>>>(m, ly, lyz, eps, out);
    sv_z_kernel<<<(NS * D_Z) / 256, 256, 0, stream>>>(m, lz, eps, out);
}